// TransformNonlinear_15075335209361
// MI455X (gfx1250) — compile-verified
//
#include <hip/hip_runtime.h>

typedef __attribute__((ext_vector_type(2))) float v2f;
typedef __attribute__((ext_vector_type(8))) float v8f;

#define BATCH 16384
#define NN    4096
#define BS    32
#define NB    (NN / BS)            // 128 blocks
#define ROWS_PER_WG 128
#define WAVES_PER_WG 8
#define THREADS (WAVES_PER_WG * 32)
#define VSTRIDE 36                 // padded LDS row stride (floats): rows 16B-aligned,
                                   // conflict-free float2 fragment reads

__global__ __launch_bounds__(THREADS)
void bdiag_wmma_kernel(const float* __restrict__ v,
                       const float* __restrict__ dx,
                       const float* __restrict__ Am,
                       const float* __restrict__ Bm,
                       const float* __restrict__ bsc,
                       float* __restrict__ out)
{
    __shared__ float lds_v[ROWS_PER_WG * VSTRIDE];   // 18.0 KB: v tile 128x32 (padded)
    __shared__ float lds_a[BS * BS];                 // 4 KB: one 32x32 A block (row-major)
    __shared__ float lds_bf[16 * 64];                // 4 KB: B fragments, [frag][lane][2]

    const int tid  = threadIdx.x;
    const int wg   = blockIdx.x;
    const int blk  = wg & (NB - 1);             // neuron block index 0..127
    const int row0 = (wg >> 7) * ROWS_PER_WG;   // batch row base

    // ---- stage A block (32x32 f32, row-major K x N) to LDS: 256 x float4, coalesced ----
    ((float4*)lds_a)[tid] = ((const float4*)(Am + (size_t)blk * (BS * BS)))[tid];

    // ---- stage v tile (128 rows x 32 cols) to LDS, 128B-coalesced per 8 lanes ----
    {
        const int r  = tid >> 3;   // 0..31
        const int cg = tid & 7;    // col group of 4 floats
        const float* gsrc = v + (size_t)(row0 + r) * NN + blk * BS + cg * 4;
        float*       ldst = lds_v + r * VSTRIDE + cg * 4;
#pragma unroll
        for (int it = 0; it < 4; ++it) {
            *(float4*)(ldst + it * 32 * VSTRIDE) =
                *(const float4*)(gsrc + (size_t)it * 32 * NN);
        }
    }
    __syncthreads();

    // ---- swizzle A block into WMMA B-fragment order: one b64 per lane per fragment ----
    // fragment f = j*8 + s (j: 16-col tile, s: K-step). Per ISA 4x16 f32 B layout:
    // lane L holds rows {k0+2*(L>>4), k0+1+2*(L>>4)} at col 16*j + (L&15), k0 = 4*s.
    {
        const int f   = tid >> 4;        // 0..15
        const int rem = tid & 15;        // lane pair index
        const int s   = f & 7;
        const int j   = f >> 3;
        float4 w;
#pragma unroll
        for (int q = 0; q < 2; ++q) {
            const int L  = rem * 2 + q;
            const int k0 = 4 * s + 2 * (L >> 4);
            const int c  = 16 * j + (L & 15);
            (&w.x)[2 * q]     = lds_a[(k0    ) * BS + c];
            (&w.x)[2 * q + 1] = lds_a[(k0 + 1) * BS + c];
        }
        *(float4*)(lds_bf + f * 64 + rem * 4) = w;
    }
    __syncthreads();

    const int lane = tid & 31;
    const int wave = tid >> 5;
    const int m    = lane & 15;    // output row-in-tile (A frag) / col-in-tile (B frag)
    const int half = lane >> 4;    // K-half selector per ISA 16x4 f32 layout

    // ---- WMMA main loop: [16x32] v-tile x [32x32] A-block, K chained in steps of 4 ----
    v8f c0 = {};
    v8f c1 = {};
    const float* vrow = lds_v + (16 * wave + m) * VSTRIDE + 2 * half;
    const float* bfp  = lds_bf + lane * 2;     // bank-conflict-free b64 per fragment
#pragma unroll
    for (int s = 0; s < 8; ++s) {
        v2f a  = *(const v2f*)(vrow + 4 * s);        // lane<16: K0,K1 ; lane>=16: K2,K3
        v2f b0 = *(const v2f*)(bfp + s * 64);        // fragment j=0, step s
        v2f b1 = *(const v2f*)(bfp + (8 + s) * 64);  // fragment j=1, step s
        c0 = __builtin_amdgcn_wmma_f32_16x16x4_f32(false, a, false, b0,
                                                   (short)0, c0, false, false);
        c1 = __builtin_amdgcn_wmma_f32_16x16x4_f32(false, a, false, b1,
                                                   (short)0, c1, false, false);
    }

    // ---- epilogue: + dx @ Bm^T + b, ReLU, coalesced store ----
    const float bias  = bsc[0];
    const int colbase = blk * BS;
    const v2f bm0 = *(const v2f*)(Bm + (size_t)(colbase + m) * 2);
    const v2f bm1 = *(const v2f*)(Bm + (size_t)(colbase + 16 + m) * 2);

#pragma unroll
    for (int i = 0; i < 8; ++i) {
        // C/D layout: VGPR i -> M = i (lanes 0-15) or i+8 (lanes 16-31), N = lane&15
        const int row = row0 + 16 * wave + i + 8 * half;
        const v2f dxv = *(const v2f*)(dx + (size_t)row * 2);
        float r0 = c0[i] + dxv.x * bm0.x + dxv.y * bm0.y + bias;
        float r1 = c1[i] + dxv.x * bm1.x + dxv.y * bm1.y + bias;
        r0 = r0 > 0.f ? r0 : 0.f;
        r1 = r1 > 0.f ? r1 : 0.f;
        float* o = out + (size_t)row * NN + colbase;
        o[m]      = r0;
        o[16 + m] = r1;
    }
}

extern "C" void kernel_launch(void* const* d_in, const int* in_sizes, int n_in,
                              void* d_out, int out_size, void* d_ws, size_t ws_size,
                              hipStream_t stream) {
    const float* v   = (const float*)d_in[0];   // [16384, 4096]
    const float* dx  = (const float*)d_in[1];   // [16384, 2]
    const float* Am  = (const float*)d_in[2];   // [128, 32, 32]
    const float* Bm  = (const float*)d_in[3];   // [4096, 2]
    const float* b   = (const float*)d_in[4];   // scalar
    float*       out = (float*)d_out;           // [16384, 4096]

    dim3 grid((BATCH / ROWS_PER_WG) * NB);      // 128 row-tiles * 128 blocks = 16384 WGs
    bdiag_wmma_kernel<<<grid, THREADS, 0, stream>>>(v, dx, Am, Bm, b, out);
}